// ICP_42425686949950
// MI455X (gfx1250) — compile-verified
//
#include <hip/hip_runtime.h>

typedef __attribute__((ext_vector_type(2))) float v2f;
typedef __attribute__((ext_vector_type(8))) float v8f;

#define BATCH 8
#define NPTS 2048
#define TOLR 1e-4f

// stats layout (floats)
#define ST_ERRNEW 0
#define ST_ERR    8
#define ST_DONE   16
#define ST_ACC    24      // per batch 16 floats: Ssrc[3], Sdst[3], Sxy[9]
#define ST_RT     152     // per batch 12 floats: R[9], t[3]

// ---------------------------------------------------------------------------
// init: bmat[b][n] = (-2x,-2y,-2z,|p|^2) from p2; err=1e9; done=0
// ---------------------------------------------------------------------------
__global__ __launch_bounds__(256) void icp_init(const float* __restrict__ p2,
                                                float* __restrict__ bmat,
                                                float* __restrict__ stats) {
    int g = blockIdx.x * 256 + threadIdx.x;            // 0..16383
    int b = g >> 11, n = g & (NPTS - 1);
    const float* q = p2 + ((size_t)(b * NPTS + n)) * 3;
    float x = q[0], y = q[1], z = q[2];
    float4 v;
    v.x = -2.f * x; v.y = -2.f * y; v.z = -2.f * z; v.w = x*x + y*y + z*z;
    ((float4*)bmat)[b * NPTS + n] = v;
    if (g < BATCH) stats[ST_ERR + g] = 1e9f;
    if (g == 0)    stats[ST_DONE] = 0.f;
}

// ---------------------------------------------------------------------------
// NN search via V_WMMA_F32_16X16X4_F32:
//   s[n][m] = |b_m|^2 - 2 a_n . b_m   (argmin over m equals argmin of d^2)
// A row = (ax,ay,az,1), B col = (-2bx,-2by,-2bz,|b|^2), K padded to 4.
// grid = BATCH*16 blocks, 8 waves/block, one 16-row tile per wave.
// ---------------------------------------------------------------------------
__global__ __launch_bounds__(256) void icp_nn(const float* __restrict__ bmat,
                                              const float* __restrict__ temppc,
                                              float* __restrict__ dmin,
                                              int* __restrict__ idxout) {
    __shared__ float4 ldsB[NPTS];                      // 32 KB of 320 KB/WGP
    int b    = blockIdx.x >> 4;
    int grp  = blockIdx.x & 15;
    int lane = threadIdx.x & 31;
    int wave = threadIdx.x >> 5;

    const float4* src = (const float4*)bmat + (size_t)b * NPTS;
    for (int i = threadIdx.x; i < NPTS; i += 256) ldsB[i] = src[i];
    __syncthreads();

    int rowbase = (grp * 8 + wave) * 16;
    int row = rowbase + (lane & 15);
    const float* tp = temppc + ((size_t)b * NPTS + row) * 3;
    float ax = tp[0], ay = tp[1], az = tp[2];
    float anrm = ax*ax + ay*ay + az*az;

    // A 16x4 f32 layout: v0 = K0 (lanes 0-15) / K2 (lanes 16-31); v1 = K1 / K3
    v2f A;
    A.x = (lane < 16) ? ax : az;
    A.y = (lane < 16) ? ay : 1.0f;

    float best[8];
    int   besti[8];
#pragma unroll
    for (int r = 0; r < 8; ++r) { best[r] = 3.4e38f; besti[r] = 0; }

#pragma unroll 2
    for (int mt = 0; mt < NPTS / 16; ++mt) {
        int col = mt * 16 + (lane & 15);
        float4 bp = ldsB[col];
        // B 4x16 f32 layout mirrors A: v0 = K0/K2, v1 = K1/K3, N = lane
        v2f Bm;
        Bm.x = (lane < 16) ? bp.x : bp.z;
        Bm.y = (lane < 16) ? bp.y : bp.w;
        v8f c = {0.f, 0.f, 0.f, 0.f, 0.f, 0.f, 0.f, 0.f};
        v8f d = __builtin_amdgcn_wmma_f32_16x16x4_f32(
            false, A, false, Bm, (short)0, c, false, false);
        // D layout: VGPR r -> row r (lanes 0-15) / row r+8 (lanes 16-31), col = lane&15
#pragma unroll
        for (int r = 0; r < 8; ++r) {
            float v = d[r];
            bool upd = v < best[r];                    // first occurrence wins across mt
            besti[r] = upd ? col : besti[r];
            best[r]  = upd ? v   : best[r];
        }
    }

    // reduce across the 16 lanes of each half (xor<16 stays within half)
#pragma unroll
    for (int off = 1; off < 16; off <<= 1) {
#pragma unroll
        for (int r = 0; r < 8; ++r) {
            float ov = __shfl_xor(best[r], off);
            int   oi = __shfl_xor(besti[r], off);
            bool take = (ov < best[r]) || (ov == best[r] && oi < besti[r]);
            best[r]  = take ? ov : best[r];
            besti[r] = take ? oi : besti[r];
        }
    }

    // rows 0..7 live in lanes 0-15 (slot r), rows 8..15 in lanes 16-31 (slot r)
    size_t obase = (size_t)b * NPTS;
    if (lane < 8) {
        int r = lane, orow = rowbase + lane;
        dmin[obase + orow]   = sqrtf(fmaxf(anrm + best[r], 1e-12f));
        idxout[obase + orow] = besti[r];
    } else if (lane >= 24) {
        int r = lane - 24, orow = rowbase + (lane - 16);
        dmin[obase + orow]   = sqrtf(fmaxf(anrm + best[r], 1e-12f));
        idxout[obase + orow] = besti[r];
    }
}

// ---------------------------------------------------------------------------
// per-batch reduction: err mean + centroids + cross-covariance sums
// mode 0: src=temppc, dst=p2 gathered by batch-7 indices (reference quirk)
// mode 1: src=p1, dst=temppc (final Kabsch), no error term
// ---------------------------------------------------------------------------
__global__ __launch_bounds__(256) void icp_reduce(const float* __restrict__ src,
                                                  const float* __restrict__ dst,
                                                  const int* __restrict__ idx7,
                                                  const float* __restrict__ dmin,
                                                  float* __restrict__ stats,
                                                  int mode) {
    int b = blockIdx.x;
    int lane = threadIdx.x & 31;
    int wave = threadIdx.x >> 5;
    float acc[16];
#pragma unroll
    for (int k = 0; k < 16; ++k) acc[k] = 0.f;

    for (int n = threadIdx.x; n < NPTS; n += 256) {
        const float* s = src + ((size_t)b * NPTS + n) * 3;
        int m = mode ? n : idx7[n];
        const float* d = dst + ((size_t)b * NPTS + m) * 3;
        float sx = s[0], sy = s[1], sz = s[2];
        float dx = d[0], dy = d[1], dz = d[2];
        if (!mode) acc[0] += dmin[(size_t)b * NPTS + n];
        acc[1] += sx; acc[2] += sy; acc[3] += sz;
        acc[4] += dx; acc[5] += dy; acc[6] += dz;
        acc[7]  += sx * dx; acc[8]  += sx * dy; acc[9]  += sx * dz;
        acc[10] += sy * dx; acc[11] += sy * dy; acc[12] += sy * dz;
        acc[13] += sz * dx; acc[14] += sz * dy; acc[15] += sz * dz;
    }

    __shared__ float sred[8][16];
#pragma unroll
    for (int off = 16; off > 0; off >>= 1)
#pragma unroll
        for (int k = 0; k < 16; ++k) acc[k] += __shfl_down(acc[k], off);
    if (lane == 0)
        for (int k = 0; k < 16; ++k) sred[wave][k] = acc[k];
    __syncthreads();
    if (threadIdx.x == 0) {
        float tot[16];
        for (int k = 0; k < 16; ++k) {
            float t = 0.f;
            for (int w = 0; w < 8; ++w) t += sred[w][k];
            tot[k] = t;
        }
        if (!mode) stats[ST_ERRNEW + b] = tot[0] * (1.f / NPTS);
        float* ab = stats + ST_ACC + b * 16;
        for (int k = 0; k < 15; ++k) ab[k] = tot[1 + k];
    }
}

// ---------------------------------------------------------------------------
// 3x3 Kabsch: SVD via Jacobi on H^T H, R = V diag(1,1,d) U^T, t = cd - R cs
// ---------------------------------------------------------------------------
__device__ inline float det3(const float* M) {
    return M[0]*(M[4]*M[8]-M[5]*M[7]) - M[1]*(M[3]*M[8]-M[5]*M[6])
         + M[2]*(M[3]*M[7]-M[4]*M[6]);
}

__device__ void kabsch3(const float* Hm, const float* cs, const float* cd,
                        float* R, float* t) {
    float S[9];
    for (int i = 0; i < 3; ++i)
        for (int j = 0; j < 3; ++j) {
            float a = 0.f;
            for (int k = 0; k < 3; ++k) a += Hm[k*3+i] * Hm[k*3+j];
            S[i*3+j] = a;
        }
    float V[9] = {1,0,0, 0,1,0, 0,0,1};
    const int PP[3] = {0,0,1}, QQ[3] = {1,2,2};
    for (int sweep = 0; sweep < 10; ++sweep) {
        for (int r = 0; r < 3; ++r) {
            int p = PP[r], q = QQ[r];
            float apq = S[p*3+q];
            if (fabsf(apq) < 1e-18f) continue;
            float app = S[p*3+p], aqq = S[q*3+q];
            float tau = (aqq - app) / (2.f * apq);
            float tt  = (tau >= 0.f ? 1.f : -1.f) / (fabsf(tau) + sqrtf(1.f + tau*tau));
            float c = 1.f / sqrtf(1.f + tt*tt), sn = tt * c;
            for (int k = 0; k < 3; ++k) {
                float skp = S[k*3+p], skq = S[k*3+q];
                S[k*3+p] = c*skp - sn*skq;  S[k*3+q] = sn*skp + c*skq;
            }
            for (int k = 0; k < 3; ++k) {
                float spk = S[p*3+k], sqk = S[q*3+k];
                S[p*3+k] = c*spk - sn*sqk;  S[q*3+k] = sn*spk + c*sqk;
            }
            for (int k = 0; k < 3; ++k) {
                float vkp = V[k*3+p], vkq = V[k*3+q];
                V[k*3+p] = c*vkp - sn*vkq;  V[k*3+q] = sn*vkp + c*vkq;
            }
        }
    }
    float lam[3] = {S[0], S[4], S[8]};
    for (int a = 0; a < 2; ++a)
        for (int bb = a + 1; bb < 3; ++bb)
            if (lam[bb] > lam[a]) {
                float tl = lam[a]; lam[a] = lam[bb]; lam[bb] = tl;
                for (int k = 0; k < 3; ++k) {
                    float tv = V[k*3+a]; V[k*3+a] = V[k*3+bb]; V[k*3+bb] = tv;
                }
            }
    float U[9];
    for (int j = 0; j < 3; ++j) {
        float sg = sqrtf(fmaxf(lam[j], 0.f));
        if (sg > 1e-10f) {
            float inv = 1.f / sg;
            for (int i = 0; i < 3; ++i) {
                float a = 0.f;
                for (int k = 0; k < 3; ++k) a += Hm[i*3+k] * V[k*3+j];
                U[i*3+j] = a * inv;
            }
        } else {                               // rank-deficient: cross of other cols
            int j0 = (j+1)%3, j1 = (j+2)%3;
            U[0*3+j] = U[1*3+j0]*U[2*3+j1] - U[2*3+j0]*U[1*3+j1];
            U[1*3+j] = U[2*3+j0]*U[0*3+j1] - U[0*3+j0]*U[2*3+j1];
            U[2*3+j] = U[0*3+j0]*U[1*3+j1] - U[1*3+j0]*U[0*3+j1];
        }
    }
    float d = (det3(U) * det3(V) < 0.f) ? -1.f : 1.f;
    for (int i = 0; i < 3; ++i)
        for (int k = 0; k < 3; ++k)
            R[i*3+k] = V[i*3+0]*U[k*3+0] + V[i*3+1]*U[k*3+1] + d*V[i*3+2]*U[k*3+2];
    for (int i = 0; i < 3; ++i)
        t[i] = cd[i] - (R[i*3+0]*cs[0] + R[i*3+1]*cs[1] + R[i*3+2]*cs[2]);
}

// mode 0: convergence bookkeeping + R,t -> stats; mode 1: final R,t -> out[B,3,4]
__global__ void icp_svd(float* __restrict__ stats, float* __restrict__ outp,
                        int iter, int mode) {
    int b = threadIdx.x;
    __shared__ float s_dn;
    if (mode == 0) {
        bool c = false;
        if (b < BATCH) {
            float en = stats[ST_ERRNEW + b], eo = stats[ST_ERR + b];
            c = fabsf((en - eo) / eo) < TOLR;
        }
        unsigned long long m = __ballot(c);
        if (b == 0) {
            bool conv = (iter > 0) && ((m & 0xffull) == 0xffull);
            float dn = (stats[ST_DONE] != 0.f || conv) ? 1.f : 0.f;
            stats[ST_DONE] = dn;
            s_dn = dn;
        }
        __syncthreads();
        if (b < BATCH && s_dn == 0.f) stats[ST_ERR + b] = stats[ST_ERRNEW + b];
    }
    if (b < BATCH) {
        const float* ab = stats + ST_ACC + b * 16;
        float invN = 1.f / NPTS;
        float cs[3] = {ab[0]*invN, ab[1]*invN, ab[2]*invN};
        float cd[3] = {ab[3]*invN, ab[4]*invN, ab[5]*invN};
        float Hm[9];
        for (int i = 0; i < 3; ++i)
            for (int j = 0; j < 3; ++j)
                Hm[i*3+j] = ab[6 + i*3 + j] - (float)NPTS * cs[i] * cd[j];
        float R[9], t[3];
        kabsch3(Hm, cs, cd, R, t);
        if (mode == 0) {
            float* rt = stats + ST_RT + b * 12;
            for (int k = 0; k < 9; ++k) rt[k] = R[k];
            rt[9] = t[0]; rt[10] = t[1]; rt[11] = t[2];
        } else {
            float* o = outp + b * 12;
            for (int i = 0; i < 3; ++i) {
                o[i*4+0] = R[i*3+0]; o[i*4+1] = R[i*3+1];
                o[i*4+2] = R[i*3+2]; o[i*4+3] = t[i];
            }
        }
    }
}

// temppc = done ? temppc : R*temppc + t
__global__ __launch_bounds__(256) void icp_xform(float* __restrict__ temppc,
                                                 const float* __restrict__ stats) {
    if (stats[ST_DONE] != 0.f) return;
    int g = blockIdx.x * 256 + threadIdx.x;
    int b = g >> 11, n = g & (NPTS - 1);
    float* p = temppc + ((size_t)b * NPTS + n) * 3;
    const float* rt = stats + ST_RT + b * 12;
    float x = p[0], y = p[1], z = p[2];
    float nx = rt[0]*x + rt[1]*y + rt[2]*z + rt[9];
    float ny = rt[3]*x + rt[4]*y + rt[5]*z + rt[10];
    float nz = rt[6]*x + rt[7]*y + rt[8]*z + rt[11];
    p[0] = nx; p[1] = ny; p[2] = nz;
}

// ---------------------------------------------------------------------------
extern "C" void kernel_launch(void* const* d_in, const int* in_sizes, int n_in,
                              void* d_out, int out_size, void* d_ws, size_t ws_size,
                              hipStream_t stream) {
    const float* p1 = (const float*)d_in[0];
    const float* p2 = (const float*)d_in[1];

    float* w      = (float*)d_ws;
    float* temppc = w;                         // 8*2048*3 = 49152 floats
    float* bmat   = w + 49152;                 // 8*2048*4 = 65536 floats
    float* dmin   = bmat + 65536;              // 16384 floats
    int*   idx    = (int*)(dmin + 16384);      // 16384 ints
    float* stats  = (float*)(idx + 16384);     // 256 floats

    icp_init<<<64, 256, 0, stream>>>(p2, bmat, stats);
    hipMemcpyAsync(temppc, p1, (size_t)BATCH * NPTS * 3 * sizeof(float),
                   hipMemcpyDeviceToDevice, stream);

    for (int i = 0; i <= 8; ++i) {             // STEPLIM+1 = 9 iterations
        icp_nn<<<BATCH * 16, 256, 0, stream>>>(bmat, temppc, dmin, idx);
        icp_reduce<<<BATCH, 256, 0, stream>>>(temppc, p2, idx + 7 * NPTS, dmin, stats, 0);
        icp_svd<<<1, 32, 0, stream>>>(stats, nullptr, i, 0);
        icp_xform<<<64, 256, 0, stream>>>(temppc, stats);
    }

    icp_reduce<<<BATCH, 256, 0, stream>>>(p1, temppc, nullptr, nullptr, stats, 1);
    icp_svd<<<1, 32, 0, stream>>>(stats, (float*)d_out, 0, 1);
}